// VectorQuantizer2_33732673142789
// MI455X (gfx1250) — compile-verified
//
#include <hip/hip_runtime.h>

// ---------------------------------------------------------------------------
// VQ-VAE vector quantizer for gfx1250 (MI455X).
//   z   : [4, 8, 64, 64] f32   (d_in[0])
//   emb : [16384, 8]     f32   (d_in[1])
// Outputs (flat, concatenated):
//   z_q  : 131072 f32 (BCHW) | loss : 1 f32 | idx : 16384 i32
//
// score[n][k] = ||e_k||^2 - 2 * z_n . e_k  via V_WMMA_F32_16X16X4_F32,
// emb staged to LDS with GLOBAL_LOAD_ASYNC_TO_LDS_B128, codebook split 4-way
// across blocks for occupancy, partials merged in a final pass.
//
// d_ws layout: e2[16384] f32 | wsv[4*16384] f32 | wsi[4*16384] i32  (~576 KB)
// ---------------------------------------------------------------------------

typedef float v2f __attribute__((ext_vector_type(2)));
typedef float v8f __attribute__((ext_vector_type(8)));

#define N_PTS    16384
#define N_CODES  16384
#define EDIM     8
#define CHUNK    512                          // codes staged in LDS at a time
#define TILES_PER_CHUNK (CHUNK / 16)          // 32
#define KSPLIT   4                            // codebook split across blockIdx.y
#define CODES_PER_SPLIT (N_CODES / KSPLIT)    // 4096
#define CHUNKS_PER_SPLIT (CODES_PER_SPLIT / CHUNK)  // 8
#define ROWS_PER_BLOCK 128                    // 8 waves * 16 rows

// ---------------------------------------------------------------------------
// Kernel 0: e2[k] = ||emb[k]||^2  (one-time, so emb staging is a pure copy)
// ---------------------------------------------------------------------------
__global__ __launch_bounds__(256) void vq_e2_kernel(
    const float* __restrict__ emb, float* __restrict__ e2g)
{
    const int k = blockIdx.x * 256 + threadIdx.x;
    const float4* r = (const float4*)(emb + k * EDIM);
    const float4 a = r[0], b = r[1];
    e2g[k] = a.x*a.x + a.y*a.y + a.z*a.z + a.w*a.w
           + b.x*b.x + b.y*b.y + b.z*b.z + b.w*b.w;
}

// ---------------------------------------------------------------------------
// Kernel 1: distances + argmin over one codebook split.
// One wave owns 16 z-rows; 8 waves share async-staged emb chunks in LDS.
// ---------------------------------------------------------------------------
__global__ __launch_bounds__(256) void vq_argmin_kernel(
    const float* __restrict__ z, const float* __restrict__ emb,
    const float* __restrict__ e2g,
    float* __restrict__ wsv, int* __restrict__ wsi)
{
    __shared__ float lds_emb[CHUNK * EDIM];   // 16 KB

    const int tid   = threadIdx.x;
    const int wave  = tid >> 5;
    const int lane  = tid & 31;
    const int half  = lane >> 4;  // 0: rows 0-7, K{0,1}; 1: rows 8-15, K{2,3}
    const int lm    = lane & 15;
    const int split = blockIdx.y;

    const int rowbase = blockIdx.x * ROWS_PER_BLOCK + wave * 16;

    // --- A tile (z rows), loaded once: zf[n][c] = z[b*32768 + c*4096 + hw]
    const int rowA = rowbase + lm;            // A row M = lane&15
    const int bA   = rowA >> 12;
    const int hwA  = rowA & 4095;
    const float* zb = z + bA * (EDIM * 4096) + hwA;
    const int k0 = half * 2;                  // lanes 0-15: K=0,1; 16-31: K=2,3
    v2f aLo, aHi;
    aLo.x = zb[(k0 + 0) * 4096];
    aLo.y = zb[(k0 + 1) * 4096];
    aHi.x = zb[(k0 + 4) * 4096];
    aHi.y = zb[(k0 + 5) * 4096];

    float best[8];
    int   bidx[8];
#pragma unroll
    for (int r = 0; r < 8; ++r) { best[r] = 3.0e38f; bidx[r] = 0; }

    for (int ch = 0; ch < CHUNKS_PER_SPLIT; ++ch) {
        const int codebase = split * CODES_PER_SPLIT + ch * CHUNK;

        // ---- async stage: 256 threads x 64 B = 16 KB, LDS offset tracks
        //      global offset via the shared INST_OFFSET field ----
        {
            const unsigned long long gaddr =
                (unsigned long long)(const void*)(emb + (codebase + tid * 2) * EDIM);
            const unsigned int ldsa =
                (unsigned int)(unsigned long long)(const void*)(lds_emb + tid * 2 * EDIM);
            asm volatile(
                "global_load_async_to_lds_b128 %0, %1, off offset:0\n\t"
                "global_load_async_to_lds_b128 %0, %1, off offset:16\n\t"
                "global_load_async_to_lds_b128 %0, %1, off offset:32\n\t"
                "global_load_async_to_lds_b128 %0, %1, off offset:48"
                :: "v"(ldsa), "v"(gaddr) : "memory");
            asm volatile("s_wait_asynccnt 0x0" ::: "memory");
        }
        __syncthreads();

        int col = codebase + lm;              // this lane's column (B col N)
        for (int t = 0; t < TILES_PER_CHUNK; ++t) {
            const int cl = t * 16 + lm;
            const float2 bLo2 = *(const float2*)&lds_emb[cl * EDIM + k0];
            const float2 bHi2 = *(const float2*)&lds_emb[cl * EDIM + k0 + 4];
            const float  e2c  = e2g[col];
            v2f bLo, bHi;
            bLo.x = bLo2.x; bLo.y = bLo2.y;
            bHi.x = bHi2.x; bHi.y = bHi2.y;

            v8f c = {};
            c = __builtin_amdgcn_wmma_f32_16x16x4_f32(
                    false, aLo, false, bLo, (short)0, c, false, false);
            c = __builtin_amdgcn_wmma_f32_16x16x4_f32(
                    false, aHi, false, bHi, (short)0, c, false, false);

#pragma unroll
            for (int r = 0; r < 8; ++r) {
                const float score = fmaf(-2.0f, c[r], e2c);
                const bool lt = (score < best[r]);
                best[r] = lt ? score : best[r];
                bidx[r] = lt ? col   : bidx[r];
            }
            col += 16;
        }
        __syncthreads();
    }

    // ---- reduce across the 16 lanes sharing each row set (tie: lower idx) --
#pragma unroll
    for (int r = 0; r < 8; ++r) {
        float v = best[r];
        int   i = bidx[r];
#pragma unroll
        for (int m = 8; m >= 1; m >>= 1) {
            const float ov = __shfl_xor(v, m, 16);
            const int   oi = __shfl_xor(i, m, 16);
            if (ov < v || (ov == v && oi < i)) { v = ov; i = oi; }
        }
        best[r] = v; bidx[r] = i;
    }
    if (lm == 0) {
        const int base = split * N_PTS + rowbase + half * 8;
#pragma unroll
        for (int r = 0; r < 8; ++r) {
            wsv[base + r] = best[r];
            wsi[base + r] = bidx[r];
        }
    }
}

// ---------------------------------------------------------------------------
// Kernel 2: merge the KSPLIT partial argmins (tie-break: smaller code index)
// ---------------------------------------------------------------------------
__global__ __launch_bounds__(256) void vq_merge_kernel(
    const float* __restrict__ wsv, const int* __restrict__ wsi,
    int* __restrict__ idx_out)
{
    const int n = blockIdx.x * 256 + threadIdx.x;
    float v = wsv[n];
    int   i = wsi[n];
#pragma unroll
    for (int s = 1; s < KSPLIT; ++s) {
        const float ov = wsv[s * N_PTS + n];
        const int   oi = wsi[s * N_PTS + n];
        if (ov < v || (ov == v && oi < i)) { v = ov; i = oi; }
    }
    idx_out[n] = i;
}

// ---------------------------------------------------------------------------
// Kernel 3: z_q gather back to BCHW (straight-through value == emb[idx])
// ---------------------------------------------------------------------------
__global__ __launch_bounds__(256) void vq_gather_kernel(
    const float* __restrict__ emb, const int* __restrict__ idx,
    float* __restrict__ zq_out)
{
    const int i = blockIdx.x * 256 + threadIdx.x;   // 131072 total
    const int c = (i >> 12) & 7;
    const int n = ((i >> 15) << 12) | (i & 4095);   // b*4096 + hw
    zq_out[i] = emb[idx[n] * EDIM + c];
}

// ---------------------------------------------------------------------------
// Kernel 4: loss = 1.25 * mean((emb[idx] - z)^2). Single block, fixed-order
// tree reduction -> bitwise deterministic across graph replays.
// ---------------------------------------------------------------------------
__global__ __launch_bounds__(1024) void vq_loss_kernel(
    const float* __restrict__ z, const float* __restrict__ emb,
    const int* __restrict__ idx, float* __restrict__ loss_out)
{
    __shared__ float red[1024];
    const int tid = threadIdx.x;
    float acc = 0.0f;
    for (int n = tid; n < N_PTS; n += 1024) {
        const int b  = n >> 12;
        const int hw = n & 4095;
        const float* zb = z + b * (EDIM * 4096) + hw;
        const float* e  = emb + idx[n] * EDIM;
#pragma unroll
        for (int c = 0; c < EDIM; ++c) {
            const float d = e[c] - zb[c * 4096];
            acc = fmaf(d, d, acc);
        }
    }
    red[tid] = acc;
    __syncthreads();
    for (int s = 512; s > 0; s >>= 1) {
        if (tid < s) red[tid] += red[tid + s];
        __syncthreads();
    }
    if (tid == 0)
        loss_out[0] = 1.25f * red[0] / (float)(N_PTS * EDIM);
}

// ---------------------------------------------------------------------------
extern "C" void kernel_launch(void* const* d_in, const int* in_sizes, int n_in,
                              void* d_out, int out_size, void* d_ws, size_t ws_size,
                              hipStream_t stream) {
    (void)in_sizes; (void)n_in; (void)out_size; (void)ws_size;
    const float* z   = (const float*)d_in[0];
    const float* emb = (const float*)d_in[1];

    float* out  = (float*)d_out;
    float* zq   = out;                    // 131072 f32
    float* loss = out + 131072;           // 1 f32
    int*   idx  = (int*)(out + 131073);   // 16384 i32

    float* e2g = (float*)d_ws;            // 16384 f32
    float* wsv = e2g + N_CODES;           // KSPLIT*16384 f32
    int*   wsi = (int*)(wsv + KSPLIT * N_PTS);  // KSPLIT*16384 i32

    vq_e2_kernel<<<N_CODES / 256, 256, 0, stream>>>(emb, e2g);
    vq_argmin_kernel<<<dim3(N_PTS / ROWS_PER_BLOCK, KSPLIT), 256, 0, stream>>>(
        z, emb, e2g, wsv, wsi);
    vq_merge_kernel<<<N_PTS / 256, 256, 0, stream>>>(wsv, wsi, idx);
    vq_gather_kernel<<<131072 / 256, 256, 0, stream>>>(emb, idx, zq);
    vq_loss_kernel<<<1, 1024, 0, stream>>>(z, emb, idx, loss);
}